// PartialTransport_41592463294657
// MI455X (gfx1250) — compile-verified
//
#include <hip/hip_runtime.h>

typedef float v2f __attribute__((ext_vector_type(2)));
typedef float v8f __attribute__((ext_vector_type(8)));

#define B_ 32
#define N_ 2048
#define D_ 512
#define K_ 128
#define R_ 129            // K_+1 rows in C/Q
#define GAMMA_ 10.0f
#define EPS_ 1e-12f
#define STEPS_ 10

// ---------- normalize target rows; write t_norm and tt[k] = ||t_norm_k||^2 ----------
__global__ void norm_t_kernel(const float* __restrict__ t, float* __restrict__ t_norm,
                              float* __restrict__ tt) {
    __shared__ float red[256];
    const int k = blockIdx.x;
    const int tid = threadIdx.x;
    const float* row = t + (size_t)k * D_;
    float x0 = row[tid];
    float x1 = row[tid + 256];
    red[tid] = x0 * x0 + x1 * x1;
    __syncthreads();
    for (int off = 128; off > 0; off >>= 1) {
        if (tid < off) red[tid] += red[tid + off];
        __syncthreads();
    }
    float s = red[0];
    float r = rsqrtf(s + EPS_);
    t_norm[(size_t)k * D_ + tid]       = x0 * r;
    t_norm[(size_t)k * D_ + tid + 256] = x1 * r;
    if (tid == 0) tt[k] = s * r * r;   // == s/(s+eps) = ||t_norm||^2
}

// ---------- per-(b,n) feature row stats: finv = rsqrt(ss+eps), ffn = ||f_norm||^2 ----------
__global__ void rowstats_f_kernel(const float* __restrict__ f, float* __restrict__ ffn,
                                  float* __restrict__ finv) {
    __shared__ float red[256];
    const int row = blockIdx.x;           // b*N + n
    const int tid = threadIdx.x;
    const float* p = f + (size_t)row * D_;
    float x0 = p[tid], x1 = p[tid + 256];
    red[tid] = x0 * x0 + x1 * x1;
    __syncthreads();
    for (int off = 128; off > 0; off >>= 1) {
        if (tid < off) red[tid] += red[tid + off];
        __syncthreads();
    }
    if (tid == 0) {
        float s = red[0];
        float r = rsqrtf(s + EPS_);
        finv[row] = r;
        ffn[row]  = s * r * r;
    }
}

// ---------- Q row 0: exp(-gamma * c_u) ----------
__global__ void row0_kernel(const float* __restrict__ c_u, float* __restrict__ Q) {
    int i = blockIdx.x * blockDim.x + threadIdx.x;   // over B*N
    int b = i / N_, n = i % N_;
    Q[(size_t)b * R_ * N_ + n] = __expf(-GAMMA_ * c_u[i]);
}

// ---------- batched GEMM (t_norm x f_raw^T) fused with cost + exp epilogue ----------
// grid = (N/128, B), block = 256 (8 waves as 4m x 2n). Wave tile: 32 k-rows x 64 n-cols
// (2 A-frags, 4 B-frags, 8 f32 accumulators -> 6 loads per 8 WMMAs).
__global__ void __launch_bounds__(256)
gemm_exp_kernel(const float* __restrict__ t_norm, const float* __restrict__ tt,
                const float* __restrict__ f, const float* __restrict__ ffn,
                const float* __restrict__ finv, float* __restrict__ Q) {
    const int b    = blockIdx.y;
    const int tid  = threadIdx.x;
    const int wave = tid >> 5;
    const int mw   = wave >> 1;              // 0..3
    const int nw   = wave & 1;               // 0..1
    const int lane = tid & 31;
    const int hl   = lane & 15;
    const int dOff = (lane >> 4) << 1;       // lanes 0-15: K=d..d+1 ; lanes 16-31: K=d+2..d+3
    const int m0   = mw * 32;
    const int nb   = blockIdx.x * 128 + nw * 64;

    const float* arow0 = t_norm + (size_t)(m0 + hl) * D_ + dOff;
    const float* arow1 = t_norm + (size_t)(m0 + 16 + hl) * D_ + dOff;
    const float* fbase = f + ((size_t)b * N_ + (size_t)(nb + hl)) * D_ + dOff;

    v8f acc[2][4] = {{v8f{}, v8f{}, v8f{}, v8f{}}, {v8f{}, v8f{}, v8f{}, v8f{}}};

    for (int d = 0; d < D_; d += 4) {
        v2f a0 = *(const v2f*)(arow0 + d);
        v2f a1 = *(const v2f*)(arow1 + d);
        v2f b0 = *(const v2f*)(fbase + d);
        v2f b1 = *(const v2f*)(fbase + (size_t)16 * D_ + d);
        v2f b2 = *(const v2f*)(fbase + (size_t)32 * D_ + d);
        v2f b3 = *(const v2f*)(fbase + (size_t)48 * D_ + d);
        acc[0][0] = __builtin_amdgcn_wmma_f32_16x16x4_f32(false, a0, false, b0, (short)0, acc[0][0], false, false);
        acc[0][1] = __builtin_amdgcn_wmma_f32_16x16x4_f32(false, a0, false, b1, (short)0, acc[0][1], false, false);
        acc[0][2] = __builtin_amdgcn_wmma_f32_16x16x4_f32(false, a0, false, b2, (short)0, acc[0][2], false, false);
        acc[0][3] = __builtin_amdgcn_wmma_f32_16x16x4_f32(false, a0, false, b3, (short)0, acc[0][3], false, false);
        acc[1][0] = __builtin_amdgcn_wmma_f32_16x16x4_f32(false, a1, false, b0, (short)0, acc[1][0], false, false);
        acc[1][1] = __builtin_amdgcn_wmma_f32_16x16x4_f32(false, a1, false, b1, (short)0, acc[1][1], false, false);
        acc[1][2] = __builtin_amdgcn_wmma_f32_16x16x4_f32(false, a1, false, b2, (short)0, acc[1][2], false, false);
        acc[1][3] = __builtin_amdgcn_wmma_f32_16x16x4_f32(false, a1, false, b3, (short)0, acc[1][3], false, false);
    }

    // Epilogue: C/D layout -> lanes 0-15 hold M=v, lanes 16-31 hold M=v+8
#pragma unroll
    for (int j = 0; j < 4; ++j) {
        const int nn = nb + j * 16 + hl;
        const int bn = b * N_ + nn;
        const float fi = finv[bn];
        const float fn = ffn[bn];
#pragma unroll
        for (int i = 0; i < 2; ++i) {
#pragma unroll
            for (int v = 0; v < 8; ++v) {
                const int m = m0 + i * 16 + ((lane < 16) ? v : v + 8);
                float dotn = acc[i][j][v] * fi;                 // t_norm . f_norm
                float cb   = tt[m] + fn - 2.0f * dotn;
                Q[((size_t)b * R_ + 1 + m) * N_ + nn] = __expf(-GAMMA_ * cb);
            }
        }
    }
}

// ---------- one-time column sums of Q0: S[b,n] = sum_r Q0[b,r,n] ----------
__global__ void colsumS_kernel(const float* __restrict__ Q, float* __restrict__ S) {
    int i = blockIdx.x * blockDim.x + threadIdx.x;   // over B*N
    int b = i / N_, n = i % N_;
    const float* Qb = Q + (size_t)b * R_ * N_ + n;
    float s = 0.0f;
    for (int r = 0; r < R_; ++r) s += Qb[(size_t)r * N_];
    S[i] = s;
}

// ---------- ALL 10 Sinkhorn steps in one kernel (diagonal-scaling form) ----------
// Q_t = diag(u) Q0 diag(v); u has only two distinct values per batch (row 0 vs rows 1..K).
// One block per batch; q0/S/v live in registers; s0/ss via block reduction.
__global__ void __launch_bounds__(1024)
sinkhorn_kernel(const float* __restrict__ Q,    // row 0 = q0
                const float* __restrict__ S,
                const float* __restrict__ mu,
                float* __restrict__ vden,       // out: v_n / (w_n*v_n*N + eps)
                float* __restrict__ u01) {      // out: per-batch {u0, u1}
    __shared__ float r0[1024];
    __shared__ float r1[1024];
    const int b = blockIdx.x;
    const int tid = threadIdx.x;
    const float q0a = Q[(size_t)b * R_ * N_ + tid];
    const float q0b = Q[(size_t)b * R_ * N_ + tid + 1024];
    const float Sa  = S[(size_t)b * N_ + tid];
    const float Sb  = S[(size_t)b * N_ + tid + 1024];
    const float ra  = Sa - q0a;   // sum of rows 1..K of Q0, column a
    const float rb  = Sb - q0b;
    float va = 1.0f, vb = 1.0f;
    float u0 = 1.0f, u1 = 1.0f;
    const float m = mu[b];

    for (int step = 0; step < STEPS_; ++step) {
        float wa = u0 * q0a + u1 * ra;            // current colsum / v
        float wb = u0 * q0b + u1 * rb;
        va = va / (wa * va * (float)N_ + EPS_);   // column normalize
        vb = vb / (wb * vb * (float)N_ + EPS_);
        r0[tid] = u0 * (q0a * va + q0b * vb);             // partial s0
        r1[tid] = u1 * (ra * va + rb * vb);               // partial ss
        __syncthreads();
        for (int off = 512; off > 0; off >>= 1) {
            if (tid < off) { r0[tid] += r0[tid + off]; r1[tid] += r1[tid + off]; }
            __syncthreads();
        }
        float s0 = r0[0], ss = r1[0];
        __syncthreads();                           // before reusing r0/r1
        u0 *= (1.0f - m) / (s0 + EPS_);
        u1 *= m / (ss + EPS_);
    }

    // fold the final column normalization of P into v
    float wa = u0 * q0a + u1 * ra;
    float wb = u0 * q0b + u1 * rb;
    vden[(size_t)b * N_ + tid]        = va / (wa * va * (float)N_ + EPS_);
    vden[(size_t)b * N_ + tid + 1024] = vb / (wb * vb * (float)N_ + EPS_);
    if (tid == 0) { u01[2 * b] = u0; u01[2 * b + 1] = u1; }
}

// ---------- final: P[b,r,n] = u_r * Q0[b,r,n] * vden[b,n] (in place) ----------
__global__ void final_kernel(float* __restrict__ Q, const float* __restrict__ vden,
                             const float* __restrict__ u01) {
    size_t i = (size_t)blockIdx.x * blockDim.x + threadIdx.x;   // over B*R*N
    int n = (int)(i % N_);
    size_t t = i / N_;
    int r = (int)(t % R_);
    int b = (int)(t / R_);
    float u = (r == 0) ? u01[2 * b] : u01[2 * b + 1];
    Q[i] = Q[i] * u * vden[(size_t)b * N_ + n];
}

extern "C" void kernel_launch(void* const* d_in, const int* in_sizes, int n_in,
                              void* d_out, int out_size, void* d_ws, size_t ws_size,
                              hipStream_t stream) {
    const float* features = (const float*)d_in[0];   // (B,N,D)
    const float* mu       = (const float*)d_in[1];   // (B,)
    const float* c_u      = (const float*)d_in[2];   // (B,N)
    const float* tsup     = (const float*)d_in[3];   // (K,D)
    float* Q = (float*)d_out;                        // (B,129,N) — Q0 buffer, becomes P

    float* w = (float*)d_ws;
    float* t_norm = w;  w += (size_t)K_ * D_;
    float* tt     = w;  w += K_;
    float* ffn    = w;  w += (size_t)B_ * N_;
    float* finv   = w;  w += (size_t)B_ * N_;
    float* S      = w;  w += (size_t)B_ * N_;
    float* vden   = w;  w += (size_t)B_ * N_;
    float* u01    = w;  w += 2 * B_;

    norm_t_kernel<<<K_, 256, 0, stream>>>(tsup, t_norm, tt);
    rowstats_f_kernel<<<B_ * N_, 256, 0, stream>>>(features, ffn, finv);
    row0_kernel<<<(B_ * N_) / 256, 256, 0, stream>>>(c_u, Q);
    gemm_exp_kernel<<<dim3(N_ / 128, B_), 256, 0, stream>>>(t_norm, tt, features, ffn, finv, Q);
    colsumS_kernel<<<(B_ * N_) / 256, 256, 0, stream>>>(Q, S);
    sinkhorn_kernel<<<B_, 1024, 0, stream>>>(Q, S, mu, vden, u01);
    final_kernel<<<(B_ * R_ * N_) / 256, 256, 0, stream>>>(Q, vden, u01);
}